// PerResidueLDDTHead_58591943852065
// MI455X (gfx1250) — compile-verified
//
#include <hip/hip_runtime.h>
#include <hip/hip_bf16.h>

// -------- CDNA5 / gfx1250 vector types ---------------------------------
typedef __attribute__((ext_vector_type(2))) float v2f;
typedef __attribute__((ext_vector_type(4))) float v4f;
typedef __attribute__((ext_vector_type(8))) float v8f;

#define N_RES  2048
#define N_ATOM 32768
#define C_S    384
#define C_OUT  50
#define T_LD   64        // padded leading dim of t = s@W+b in workspace

// workspace layout (floats):
//   t  : [2048 x 64]            at ws + 0           (512 KB)
//   Wt : [64 x 384] transposed  at ws + 2048*64     ( 96 KB)  Wt[n][k] = n<50 ? W[k][n] : 0
//   bp : [64]                   at ws + 2048*64 + 64*384
#define WS_T   0
#define WS_WT  (N_RES * T_LD)
#define WS_BP  (WS_WT + T_LD * C_S)

// ========================================================================
// Phase 0: pad + transpose W, pad bias. Tiny (24K elements).
// ========================================================================
__global__ __launch_bounds__(256)
void phase0_pad(const float* __restrict__ W,
                const float* __restrict__ bias,
                float* __restrict__ Wt,
                float* __restrict__ bp)
{
    const int i = blockIdx.x * 256 + threadIdx.x;    // 0 .. 64*384-1
    const int n = i / C_S;
    const int k = i - n * C_S;
    Wt[i] = (n < C_OUT) ? W[(size_t)k * C_OUT + n] : 0.0f;
    if (i < T_LD) bp[i] = (i < C_OUT) ? bias[i] : 0.0f;
}

// ========================================================================
// Phase 1: t[2048 x 64] = s[2048 x 384] @ W + b   (cols>=50 zero-weight junk)
// fp32 WMMA 16x16x4. One wave per 16x16 output tile; 512 tiles.
// VGPR layouts (ISA 7.12.2):
//   A (16x4 f32): lane<16 -> M=lane, v0=K0,v1=K1 ; lane>=16 -> M=lane-16, v0=K2,v1=K3
//   B (4x16 f32): v0: lanes0-15 K=0 / lanes16-31 K=2 ; v1: K=1 / K=3 (N = lane&15)
//   C/D (16x16 f32): VGPR i: lanes0-15 M=i, lanes16-31 M=i+8, N = lane&15
// Both A and B fragments are unconditional global_load_b64 (Wt is transposed,
// so a lane's two K values are contiguous). No EXEC games around the WMMA.
// ========================================================================
__global__ __launch_bounds__(128)
void phase1_sW_wmma(const float* __restrict__ s,
                    const float* __restrict__ Wt,
                    const float* __restrict__ bp,
                    float* __restrict__ t)
{
    const int wid   = blockIdx.x * 4 + (threadIdx.x >> 5);   // 0..511
    const int mtile = wid >> 2;                              // 0..127
    const int ntile = wid & 3;                               // 0..3
    const int lane  = threadIdx.x & 31;
    const int half  = lane >> 4;                             // 0 or 1
    const int l15   = lane & 15;

    const float* arow = s  + (size_t)(mtile * 16 + l15) * C_S;  // A row this lane owns
    const float* bcol = Wt + (size_t)(ntile * 16 + l15) * C_S;  // B column this lane owns
    const int koff = half * 2;                                  // K pair owned by lane half

    v8f acc = {};

    for (int k0 = 0; k0 < C_S; k0 += 4) {
        v2f a  = *(const v2f*)(arow + k0 + koff);   // s[m][ka], s[m][ka+1]
        v2f bm = *(const v2f*)(bcol + k0 + koff);   // Wt[n][ka], Wt[n][ka+1]
        acc = __builtin_amdgcn_wmma_f32_16x16x4_f32(
            /*neg_a=*/false, a, /*neg_b=*/false, bm,
            /*c_mod=*/(short)0, acc, /*reuse_a=*/false, /*reuse_b=*/false);
    }

    const float bv = bp[ntile * 16 + l15];

#pragma unroll
    for (int i = 0; i < 8; ++i) {
        const int r = mtile * 16 + i + half * 8;
        t[(size_t)r * T_LD + ntile * 16 + l15] = acc[i] + bv;
    }
}

// ========================================================================
// Phase 2 (bandwidth-dominant, ~256 MB @ 23.3 TB/s): one wave per atom.
//  - stream the 2048-float one-hot row with NT float4 loads
//  - idx = row . iota  (exact: single 1.0 at column idx, idx <= 2047)
//  - wave32 shfl-xor reduction, then gather t[idx][0..49] -> out[atom][0..49]
// ========================================================================
__global__ __launch_bounds__(256)
void phase2_scan_gather(const float* __restrict__ onehot,
                        const float* __restrict__ t,
                        float* __restrict__ out)
{
    const int lane = threadIdx.x & 31;
    const int atom = blockIdx.x * 8 + (threadIdx.x >> 5);

    const v4f* row = (const v4f*)(onehot + (size_t)atom * N_RES);

    float sum = 0.0f;
#pragma unroll
    for (int j = 0; j < 16; ++j) {
        const int e = j * 32 + lane;                       // float4 index, coalesced
        v4f v = __builtin_nontemporal_load(row + e);       // streaming: NT hint
        const float c0 = (float)(e * 4);
        sum += v.x * c0 + v.y * (c0 + 1.0f) + v.z * (c0 + 2.0f) + v.w * (c0 + 3.0f);
    }

    // wave32 reduction
#pragma unroll
    for (int off = 16; off > 0; off >>= 1)
        sum += __shfl_xor(sum, off, 32);

    const int idx = __float2int_rn(sum);                   // exact integer in fp32

    const float* trow = t + (size_t)idx * T_LD;            // L2-resident (512 KB)
    float*       orow = out + (size_t)atom * C_OUT;

    orow[lane] = trow[lane];                               // cols 0..31
    if (lane < C_OUT - 32)                                 // cols 32..49
        orow[32 + lane] = trow[32 + lane];
}

// ========================================================================
extern "C" void kernel_launch(void* const* d_in, const int* in_sizes, int n_in,
                              void* d_out, int out_size, void* d_ws, size_t ws_size,
                              hipStream_t stream)
{
    const float* s      = (const float*)d_in[0];   // [2048, 384]
    const float* onehot = (const float*)d_in[1];   // [32768, 2048]
    const float* W      = (const float*)d_in[2];   // [384, 50]
    const float* bias   = (const float*)d_in[3];   // [50]
    float*       out    = (float*)d_out;           // [32768, 50]

    float* ws = (float*)d_ws;
    float* t  = ws + WS_T;                         // [2048, 64]
    float* Wt = ws + WS_WT;                        // [64, 384]
    float* bp = ws + WS_BP;                        // [64]

    // Phase 0: pad/transpose W and bias into scratch (24576 threads)
    phase0_pad<<<dim3((T_LD * C_S) / 256), dim3(256), 0, stream>>>(W, bias, Wt, bp);

    // Phase 1: 512 tiles = 512 waves = 128 blocks x 4 waves, fp32 WMMA GEMM -> t
    phase1_sW_wmma<<<dim3(128), dim3(128), 0, stream>>>(s, Wt, bp, t);

    // Phase 2: 32768 waves (4096 blocks x 256 threads), stream + gather
    phase2_scan_gather<<<dim3(N_ATOM / 8), dim3(256), 0, stream>>>(onehot, t, out);
}